// HGT_87471303950601
// MI455X (gfx1250) — compile-verified
//
#include <hip/hip_runtime.h>
#include <math.h>

// ---------------- constants ----------------
#define N_P 100000
#define N_D 1000
#define N_S 5000
#define NTOT (N_P + N_D + N_S)
#define C 64
#define H 4
#define DH 16
#define LAYERS 2

typedef __attribute__((ext_vector_type(2))) float v2f;
typedef __attribute__((ext_vector_type(8))) float v8f;

// order-preserving float<->uint encoding for atomic max over floats
__device__ __forceinline__ unsigned encf(float f) {
    unsigned u = __float_as_uint(f);
    return (u & 0x80000000u) ? ~u : (u | 0x80000000u);
}
__device__ __forceinline__ float decf(unsigned u) {
    return (u & 0x80000000u) ? __uint_as_float(u ^ 0x80000000u) : __uint_as_float(~u);
}

// ---------------- generic fill ----------------
__global__ void hgt_fill_u32(unsigned* __restrict__ p, unsigned v, int n) {
    int i = blockIdx.x * 256 + threadIdx.x;
    if (i < n) p[i] = v;
}

// ---------------- fuse relation weights ----------------
// Wkr[k, h*16+e] = sum_d Wk[k, h*16+d] * arel[h,d,e]   (and same for Wv/mrel)
// bkr[h*16+e]    = sum_d bk[h*16+d]    * arel[h,d,e]
__global__ void hgt_fuse_rel(const float* __restrict__ Wk, const float* __restrict__ bk,
                             const float* __restrict__ arel,
                             const float* __restrict__ Wv, const float* __restrict__ bv,
                             const float* __restrict__ mrel,
                             float* __restrict__ Wkr, float* __restrict__ bkr,
                             float* __restrict__ Wvr, float* __restrict__ bvr) {
    int tid = blockIdx.x * 256 + threadIdx.x;
    if (tid >= 64 * 64) return;
    int k = tid >> 6, j = tid & 63;
    int h = j >> 4, e = j & 15;
    float sk = 0.f, sv = 0.f;
#pragma unroll
    for (int d = 0; d < 16; ++d) {
        float ar = arel[h * 256 + d * 16 + e];
        float mr = mrel[h * 256 + d * 16 + e];
        sk += Wk[k * 64 + h * 16 + d] * ar;
        sv += Wv[k * 64 + h * 16 + d] * mr;
    }
    Wkr[k * 64 + j] = sk;
    Wvr[k * 64 + j] = sv;
    if (k == 0) {
        float bsk = 0.f, bsv = 0.f;
#pragma unroll
        for (int d = 0; d < 16; ++d) {
            bsk += bk[h * 16 + d] * arel[h * 256 + d * 16 + e];
            bsv += bv[h * 16 + d] * mrel[h * 256 + d * 16 + e];
        }
        bkr[j] = bsk;
        bvr[j] = bsv;
    }
}

// ---------------- WMMA GEMM: D[N,64] = op(A[N,K]) @ W[K,64] + bias ----------------
// mode 0: D = A @ W + b
// mode 1: D = s * (gelu(A) @ W + b) + (1-s) * Xold    (s = sigmoid(*skipPtr))
// 256 threads = 8 waves; each wave one 16x16 tile; block covers 32 rows x 64 cols.
__global__ __launch_bounds__(256) void hgt_gemm64(
    const float* __restrict__ A, const float* __restrict__ W,
    const float* __restrict__ bias, const float* __restrict__ Xold,
    const float* __restrict__ skipPtr, float* __restrict__ D,
    int N, int K, int mode) {
    __shared__ float W_s[128 * 64];
    __shared__ float A_s[32 * 128];
    const int tid = threadIdx.x;
    const int rowBase = blockIdx.x * 32;

    for (int i = tid; i < K * 64; i += 256) W_s[i] = W[i];
    for (int i = tid; i < 32 * K; i += 256) {
        int r = i / K, k = i - r * K;
        int gr = rowBase + r;
        float v = (gr < N) ? A[(size_t)gr * K + k] : 0.f;
        if (mode == 1) v = 0.5f * v * (1.f + erff(v * 0.70710678118654752f));  // exact GELU
        A_s[i] = v;
    }
    __syncthreads();

    const int wid = tid >> 5, lane = tid & 31;
    const int lrow = (wid >> 2) * 16;   // 0 or 16 within block tile
    const int col0 = (wid & 3) * 16;    // 0,16,32,48
    const int ln = lane & 15, hi = lane >> 4;

    float bval = bias[col0 + ln];
    v8f c;
#pragma unroll
    for (int i = 0; i < 8; ++i) c[i] = bval;

    for (int k0 = 0; k0 < K; k0 += 4) {
        int ka = k0 + (hi ? 2 : 0);
        v2f a, b;
        a.x = A_s[(lrow + ln) * K + ka];
        a.y = A_s[(lrow + ln) * K + ka + 1];
        b.x = W_s[ka * 64 + col0 + ln];
        b.y = W_s[(ka + 1) * 64 + col0 + ln];
        // D = A(16x4,f32) x B(4x16,f32) + C  -> v_wmma_f32_16x16x4_f32
        c = __builtin_amdgcn_wmma_f32_16x16x4_f32(false, a, false, b, (short)0, c,
                                                  false, false);
    }

    float s = 0.f;
    if (mode == 1) s = 1.f / (1.f + expf(-skipPtr[0]));
#pragma unroll
    for (int i = 0; i < 8; ++i) {
        int gr = rowBase + lrow + (hi ? 8 : 0) + i;
        int gc = col0 + ln;
        if (gr < N) {
            float v = c[i];
            if (mode == 1) v = s * v + (1.f - s) * Xold[(size_t)gr * 64 + gc];
            D[(size_t)gr * 64 + gc] = v;
        }
    }
}

// ---------------- edge pass 1: logits + segment max ----------------
__global__ void hgt_edge_alpha(const int* __restrict__ ei, int E,
                               const float* __restrict__ Q, int dstBase,
                               const float* __restrict__ krel,
                               const float* __restrict__ prel,
                               float* __restrict__ alpha,
                               unsigned* __restrict__ segmax) {
    int idx = blockIdx.x * 256 + threadIdx.x;
    if (idx >= E * 4) return;
    int e = idx >> 2, h = idx & 3;
    int src = ei[e], dst = ei[E + e];
    const float4* qp = (const float4*)(Q + ((size_t)(dstBase + dst)) * 64 + h * 16);
    const float4* kp = (const float4*)(krel + (size_t)src * 64 + h * 16);
    float acc = 0.f;
#pragma unroll
    for (int i = 0; i < 4; ++i) {
        float4 q = qp[i], k = kp[i];
        acc += q.x * k.x + q.y * k.y + q.z * k.z + q.w * k.w;
    }
    float al = acc * prel[h] * 0.25f;  // 1/sqrt(DH) = 0.25
    alpha[idx] = al;
    atomicMax(&segmax[dst * 4 + h], encf(al));
}

// ---------------- edge pass 2: segment sum of exp ----------------
__global__ void hgt_edge_expsum(const int* __restrict__ ei, int E,
                                const float* __restrict__ alpha,
                                const unsigned* __restrict__ segmax,
                                float* __restrict__ segsum) {
    int idx = blockIdx.x * 256 + threadIdx.x;
    if (idx >= E * 4) return;
    int e = idx >> 2, h = idx & 3;
    int dst = ei[E + e];
    float mx = decf(segmax[dst * 4 + h]);
    atomicAdd(&segsum[dst * 4 + h], expf(alpha[idx] - mx));
}

// ---------------- edge pass 3: weighted scatter of v_rel ----------------
__global__ void hgt_edge_scatter(const int* __restrict__ ei, int E,
                                 const float* __restrict__ alpha,
                                 const unsigned* __restrict__ segmax,
                                 const float* __restrict__ segsum,
                                 const float* __restrict__ vrel,
                                 float* __restrict__ Out, int dstBase) {
    int idx = blockIdx.x * 256 + threadIdx.x;
    if (idx >= E * 64) return;
    int e = idx >> 6, cc = idx & 63, h = cc >> 4;
    int src = ei[e], dst = ei[E + e];
    float mx = decf(segmax[dst * 4 + h]);
    float w = expf(alpha[e * 4 + h] - mx) / (segsum[dst * 4 + h] + 1e-16f);
    atomicAdd(&Out[((size_t)(dstBase + dst)) * 64 + cc],
              vrel[(size_t)src * 64 + cc] * w);
}

// ---------------- host orchestration ----------------
static inline int cdiv(int a, int b) { return (a + b - 1) / b; }

extern "C" void kernel_launch(void* const* d_in, const int* in_sizes, int n_in,
                              void* d_out, int out_size, void* d_ws, size_t ws_size,
                              hipStream_t stream) {
    const float* x_p = (const float*)d_in[0];
    const float* x_d = (const float*)d_in[1];
    const float* x_s = (const float*)d_in[2];
    const float* Win_p = (const float*)d_in[3];
    const float* bin_p = (const float*)d_in[4];
    const float* Win_d = (const float*)d_in[5];
    const float* bin_d = (const float*)d_in[6];
    const float* Win_s = (const float*)d_in[7];
    const float* bin_s = (const float*)d_in[8];
    const float* Wk = (const float*)d_in[9];
    const float* bk = (const float*)d_in[10];
    const float* Wq = (const float*)d_in[11];
    const float* bq = (const float*)d_in[12];
    const float* Wv = (const float*)d_in[13];
    const float* bv = (const float*)d_in[14];
    const float* Wa = (const float*)d_in[15];
    const float* ba = (const float*)d_in[16];
    const float* a_rel = (const float*)d_in[17];
    const float* m_rel = (const float*)d_in[18];
    const float* p_rel = (const float*)d_in[19];
    const float* skip = (const float*)d_in[20];
    const int* ei_pd = (const int*)d_in[21];
    const int* ei_dp = (const int*)d_in[22];
    const int* ei_ds = (const int*)d_in[23];
    const int* ei_ps = (const int*)d_in[24];

    // workspace carve-up (floats)
    float* ws = (float*)d_ws;
    float* XS = ws;                              // NTOT*64
    float* Qb = XS + (size_t)NTOT * 64;          // NTOT*64
    float* OUTb = Qb + (size_t)NTOT * 64;        // NTOT*64
    float* KREL = OUTb + (size_t)NTOT * 64;      // N_P*64 (max src)
    float* VREL = KREL + (size_t)N_P * 64;       // N_P*64
    float* ALPH = VREL + (size_t)N_P * 64;       // 300000*4 (max E*H)
    unsigned* SMAX = (unsigned*)(ALPH + (size_t)300000 * 4);  // N_P*4
    float* SSUM = (float*)(SMAX + (size_t)N_P * 4);           // N_P*4
    float* WKR = SSUM + (size_t)N_P * 4;         // 4096
    float* BKR = WKR + 4096;                     // 64
    float* WVR = BKR + 64;                       // 4096
    float* BVR = WVR + 4096;                     // 64
    (void)ws_size; (void)n_in; (void)in_sizes; (void)out_size;

    const int Ns[3] = {N_P, N_D, N_S};
    const int base[3] = {0, N_P, N_P + N_D};

    // ---- input projections into XS ----
    hgt_gemm64<<<cdiv(N_P, 32), 256, 0, stream>>>(x_p, Win_p, bin_p, nullptr, nullptr,
                                                  XS + (size_t)base[0] * 64, N_P, 64, 0);
    hgt_gemm64<<<cdiv(N_D, 32), 256, 0, stream>>>(x_d, Win_d, bin_d, nullptr, nullptr,
                                                  XS + (size_t)base[1] * 64, N_D, 128, 0);
    hgt_gemm64<<<cdiv(N_S, 32), 256, 0, stream>>>(x_s, Win_s, bin_s, nullptr, nullptr,
                                                  XS + (size_t)base[2] * 64, N_S, 64, 0);

    struct Rel { int st, dt, E; const int* ei; };
    const Rel rels[4] = {{0, 1, 300000, ei_pd},
                         {1, 0, 300000, ei_dp},
                         {1, 2, 100000, ei_ds},
                         {0, 2, 100000, ei_ps}};

    for (int l = 0; l < LAYERS; ++l) {
        // ---- Q projections for all node types ----
        for (int t = 0; t < 3; ++t) {
            hgt_gemm64<<<cdiv(Ns[t], 32), 256, 0, stream>>>(
                XS + (size_t)base[t] * 64, Wq + (size_t)(l * 3 + t) * 4096,
                bq + (size_t)(l * 3 + t) * 64, nullptr, nullptr,
                Qb + (size_t)base[t] * 64, Ns[t], 64, 0);
        }
        // ---- zero the output accumulator ----
        {
            int n = NTOT * 64;
            hgt_fill_u32<<<cdiv(n, 256), 256, 0, stream>>>((unsigned*)OUTb, 0u, n);
        }
        // ---- relations ----
        for (int r = 0; r < 4; ++r) {
            const Rel& R = rels[r];
            hgt_fuse_rel<<<16, 256, 0, stream>>>(
                Wk + (size_t)(l * 3 + R.st) * 4096, bk + (size_t)(l * 3 + R.st) * 64,
                a_rel + (size_t)(l * 4 + r) * 1024,
                Wv + (size_t)(l * 3 + R.st) * 4096, bv + (size_t)(l * 3 + R.st) * 64,
                m_rel + (size_t)(l * 4 + r) * 1024, WKR, BKR, WVR, BVR);
            // k_rel / v_rel via fused weights (block-diag a_rel folded into Wk/Wv)
            hgt_gemm64<<<cdiv(Ns[R.st], 32), 256, 0, stream>>>(
                XS + (size_t)base[R.st] * 64, WKR, BKR, nullptr, nullptr, KREL,
                Ns[R.st], 64, 0);
            hgt_gemm64<<<cdiv(Ns[R.st], 32), 256, 0, stream>>>(
                XS + (size_t)base[R.st] * 64, WVR, BVR, nullptr, nullptr, VREL,
                Ns[R.st], 64, 0);
            // segment softmax buffers
            int nseg = Ns[R.dt] * 4;
            hgt_fill_u32<<<cdiv(nseg, 256), 256, 0, stream>>>(SMAX, 0x007FFFFFu, nseg); // enc(-inf)
            hgt_fill_u32<<<cdiv(nseg, 256), 256, 0, stream>>>((unsigned*)SSUM, 0u, nseg);
            // edge passes
            int n1 = R.E * 4;
            hgt_edge_alpha<<<cdiv(n1, 256), 256, 0, stream>>>(
                R.ei, R.E, Qb, base[R.dt], KREL, p_rel + (size_t)(l * 4 + r) * 4,
                ALPH, SMAX);
            hgt_edge_expsum<<<cdiv(n1, 256), 256, 0, stream>>>(R.ei, R.E, ALPH, SMAX, SSUM);
            int n3 = R.E * 64;
            hgt_edge_scatter<<<cdiv(n3, 256), 256, 0, stream>>>(
                R.ei, R.E, ALPH, SMAX, SSUM, VREL, OUTb, base[R.dt]);
        }
        // ---- combine: new_x = s*(gelu(out)@Wa + ba) + (1-s)*x ----
        float* dst = (l == LAYERS - 1) ? (float*)d_out : XS;
        for (int t = 0; t < 3; ++t) {
            hgt_gemm64<<<cdiv(Ns[t], 32), 256, 0, stream>>>(
                OUTb + (size_t)base[t] * 64, Wa + (size_t)(l * 3 + t) * 4096,
                ba + (size_t)(l * 3 + t) * 64, XS + (size_t)base[t] * 64,
                skip + (size_t)(l * 3 + t), dst + (size_t)base[t] * 64, Ns[t], 64, 1);
        }
    }
}